// CrossEntropyLoss_LSR_51548197487170
// MI455X (gfx1250) — compile-verified
//
#include <hip/hip_runtime.h>
#include <hip/hip_bf16.h>

// ---------------------------------------------------------------------------
// CrossEntropyLoss with label smoothing (SMOOTHING=0.2, C=3, T=5, FILLUP=-100)
// loss(b,t) = valid * ( lse - (0.2/3)*(p0+p1+p2) - 0.8*p_label )
// output    = mean over B of sum over t.
// Memory-bound: ~168MB read once -> stage predictions via CDNA5 async
// global->LDS b128 copies (contiguous 512B/wave), compute from LDS.
// ---------------------------------------------------------------------------

#define CE_BLOCK   256
#define CE_TILE    1024                 // samples per tile
#define CE_TILE_F  (CE_TILE * 15)       // floats per tile  (15360)
#define CE_SLOTS_PER_THREAD 15          // 15360*4B / 16B / 256 threads

#if defined(__HIP_DEVICE_COMPILE__) && defined(__gfx1250__) && \
    __has_builtin(__builtin_amdgcn_global_load_async_to_lds_b128)
#define CE_HAS_ASYNC 1
#else
#define CE_HAS_ASYNC 0
#endif

typedef int ce_v4i __attribute__((vector_size(16)));
typedef __attribute__((address_space(1))) ce_v4i* ce_gptr;   // global (AS1)
typedef __attribute__((address_space(3))) ce_v4i* ce_lptr;   // LDS   (AS3)

__device__ __forceinline__ void ce_async_wait_all() {
#if CE_HAS_ASYNC
#if __has_builtin(__builtin_amdgcn_s_wait_asynccnt)
    __builtin_amdgcn_s_wait_asynccnt(0);
#else
    asm volatile("s_wait_asynccnt 0" ::: "memory");
#endif
#endif
}

__global__ __launch_bounds__(CE_BLOCK)
void ce_lsr_partial_kernel(const float* __restrict__ pred,
                           const int*  __restrict__ lab,
                           float* __restrict__ partials,
                           long long B, int numTiles)
{
    __shared__ float spred[CE_TILE_F];   // 60 KB
    __shared__ float sred[CE_BLOCK];

    const int tid = threadIdx.x;
    float acc = 0.0f;

    const size_t predBytes = (size_t)B * 60u;
    const size_t clampMax  = (predBytes - 16u) & ~(size_t)15u;  // last aligned 16B chunk

    for (int tile = blockIdx.x; tile < numTiles; tile += gridDim.x) {
        const size_t tileByte = (size_t)tile * (size_t)(CE_TILE * 60);

        // ---- stage predictions tile into LDS (contiguous 16B slots) ----
#pragma unroll
        for (int t = 0; t < CE_SLOTS_PER_THREAD; ++t) {
            const int k = tid + (t << 8);               // slot 0..3839
            size_t go = tileByte + ((size_t)k << 4);
            if (go > clampMax) go = clampMax;            // tail clamp (masked later)
            char* src = (char*)pred + go;                // const dropped: builtin wants non-const
            char* dst = (char*)spred + ((size_t)k << 4);
#if CE_HAS_ASYNC
            __builtin_amdgcn_global_load_async_to_lds_b128(
                (ce_gptr)src, (ce_lptr)dst, /*offset=*/0, /*cpol=*/0);
#else
            *(float4*)dst = *(const float4*)src;
#endif
        }
        ce_async_wait_all();
        __syncthreads();

        // ---- compute: 4 samples per thread, LDS stride 15 (bank-conflict-free) ----
        const long long baseS = (long long)tile << 10;
#pragma unroll
        for (int q = 0; q < 4; ++q) {
            const int sl = tid + (q << 8);
            const long long b = baseS + sl;
            if (b < B) {
                const float* p  = &spred[sl * 15];
                const int*   lb = lab + b * 5;
#pragma unroll
                for (int t = 0; t < 5; ++t) {
                    const int l = lb[t];
                    if (l != -100) {
                        const float p0 = p[t], p1 = p[5 + t], p2 = p[10 + t];
                        const float m  = fmaxf(p0, fmaxf(p1, p2));
                        const float s  = __expf(p0 - m) + __expf(p1 - m) + __expf(p2 - m);
                        const float lse = m + __logf(s);
                        const float pl = (l == 0) ? p0 : ((l == 1) ? p1 : p2);
                        acc += lse - 0.06666666666666667f * (p0 + p1 + p2) - 0.8f * pl;
                    }
                }
            }
        }
        __syncthreads();   // protect spred before next tile refill
    }

    // ---- deterministic block tree reduction ----
    sred[tid] = acc;
    __syncthreads();
#pragma unroll
    for (int off = CE_BLOCK >> 1; off > 0; off >>= 1) {
        if (tid < off) sred[tid] += sred[tid + off];
        __syncthreads();
    }
    if (tid == 0) partials[blockIdx.x] = sred[0];
}

__global__ __launch_bounds__(CE_BLOCK)
void ce_lsr_final_kernel(const float* __restrict__ partials,
                         float* __restrict__ out, int n, float invB)
{
    __shared__ float sred[CE_BLOCK];
    const int tid = threadIdx.x;
    float a = 0.0f;
    for (int j = tid; j < n; j += CE_BLOCK) a += partials[j];  // fixed order
    sred[tid] = a;
    __syncthreads();
#pragma unroll
    for (int off = CE_BLOCK >> 1; off > 0; off >>= 1) {
        if (tid < off) sred[tid] += sred[tid + off];
        __syncthreads();
    }
    if (tid == 0) out[0] = sred[0] * invB;
}

extern "C" void kernel_launch(void* const* d_in, const int* in_sizes, int n_in,
                              void* d_out, int out_size, void* d_ws, size_t ws_size,
                              hipStream_t stream) {
    const float* pred = (const float*)d_in[0];   // [B, 3, 5] f32
    const int*   lab  = (const int*)d_in[1];     // [B, 5] int32 (jax default x32)

    const long long B = (long long)in_sizes[1] / 5;
    int numTiles = (int)((B + (CE_TILE - 1)) / CE_TILE);
    if (numTiles < 1) numTiles = 1;
    int grid = numTiles < 2048 ? numTiles : 2048;

    float* partials = (float*)d_ws;              // grid floats (<= 8KB)

    ce_lsr_partial_kernel<<<grid, CE_BLOCK, 0, stream>>>(pred, lab, partials, B, numTiles);
    ce_lsr_final_kernel<<<1, CE_BLOCK, 0, stream>>>(partials, (float*)d_out, grid,
                                                    1.0f / (float)B);
}